// LogitConstraintEnforcer_16862041604789
// MI455X (gfx1250) — compile-verified
//
#include <hip/hip_runtime.h>
#include <hip/hip_bf16.h>
#include <stdint.h>

// Problem constants (match reference)
#define B_       128
#define V_       128000
#define T_       4096
#define R_       64

// Tiling: 4 vocab slices per row -> 512 workgroups
#define SLICES   4
#define VS       (V_ / SLICES)      // 32000 tokens per slice
#define VS4      (VS / 4)           // 8000 float4 / packed-byte words
#define NTHREADS 1024

// LDS layout: [0,VS) byte histogram | [VS, VS+T*4) token staging | [.., +VS) forbidden-mask slice
#define STAGE_OFF   VS
#define FB_OFF      (VS + T_ * 4)
#define SMEM_BYTES  (VS + T_ * 4 + VS)   // 32000 + 16384 + 32000 = 80384 B

typedef unsigned uint32x4 __attribute__((ext_vector_type(4)));
typedef unsigned uint32x8 __attribute__((ext_vector_type(8)));

__device__ __forceinline__ float apply_penalty(float o) {
    // exact reference semantics: out/1.2 if >0 else out*1.2
    return (o > 0.0f) ? (o / 1.2f) : (o * 1.2f);
}

__global__ __launch_bounds__(NTHREADS)
void logit_constraint_kernel(const float* __restrict__ logits,
                             const int* __restrict__ generated,
                             const unsigned char* __restrict__ forbidden,
                             const int* __restrict__ required,
                             float* __restrict__ out) {
    extern __shared__ unsigned char smem[];
    unsigned*      histW = (unsigned*)smem;              // VS bytes of counts, packed 4/word
    int*           stage = (int*)(smem + STAGE_OFF);     // T_ staged tokens (async DMA target)
    unsigned char* fbL   = smem + FB_OFF;                // staged forbidden-mask slice (TDM target)

    const int s   = blockIdx.x;                 // vocab slice
    const int b   = blockIdx.y;                 // batch row
    const int tid = threadIdx.x;
    const int vLo = s * VS;

    // ---- Phase 0: wave 0 issues a TDM DMA of the forbidden-mask slice ----
    // (32000 contiguous bytes global -> LDS; 1-D tensor descriptor, ISA 8.3/8.4)
    if (tid < 32) {
        unsigned long long ga = (unsigned long long)(size_t)(const void*)(forbidden + vLo);
        unsigned lds_fb = (unsigned)(size_t)(const void*)fbL;   // LDS byte offset (low 32 bits)
        uint32x4 g0;
        g0.x = 1u;                                              // count=1 (valid), user mode
        g0.y = lds_fb;                                          // lds_addr
        g0.z = (unsigned)ga;                                    // global_addr[31:0]
        g0.w = (unsigned)((ga >> 32) & 0x01FFFFFFu) | (2u << 30); // global_addr[56:32] | type=2
        uint32x8 g1;
        g1.s0 = 0u << 16;                                       // wg_mask=0, data_size=0 (1B)
        g1.s1 = ((unsigned)VS & 0xFFFFu) << 16;                 // tensor_dim0[15:0]
        g1.s2 = ((unsigned)VS >> 16) | (1u << 16);              // tensor_dim0[31:16] | tensor_dim1=1
        g1.s3 = ((unsigned)VS & 0xFFFFu) << 16;                 // tile_dim0 = VS
        g1.s4 = 0u;                                             // tile_dim1/2 unused
        g1.s5 = (unsigned)VS;                                   // tensor_dim0_stride[31:0]
        g1.s6 = 0u;
        g1.s7 = 0u;
        asm volatile("tensor_load_to_lds %0, %1" :: "s"(g0), "s"(g1) : "memory");
    }

    // ---- Phase A: zero the byte-histogram (overlaps with TDM) -----------
    for (int w = tid; w < VS4; w += NTHREADS)
        histW[w] = 0u;

    // ---- Phase B: async-copy this row's 4096 tokens (16 KB) global->LDS -
    // Each lane copies the exact 16 bytes it consumes in Phase C, so a
    // per-wave s_wait_asynccnt 0 is a sufficient dependency.
    {
        unsigned lds_off = (unsigned)(size_t)(const void*)(stage + 4 * tid);
        unsigned long long gaddr =
            (unsigned long long)(size_t)(const void*)(generated + (size_t)b * T_ + 4 * tid);
        asm volatile("global_load_async_to_lds_b128 %0, %1, off"
                     :: "v"(lds_off), "v"(gaddr) : "memory");
    }

    __syncthreads();                                   // histogram zeroing visible
    asm volatile("s_wait_asynccnt 0" ::: "memory");    // own async copy done

    // ---- Phase C: byte histogram of tokens falling in this slice --------
    {
        int4 tk = ((const int4*)stage)[tid];           // ds_load_b128
        #pragma unroll
        for (int k = 0; k < 4; ++k) {
            int tok = (&tk.x)[k];
            unsigned v = (unsigned)(tok - vLo);
            if (v < (unsigned)VS)
                atomicAdd(&histW[v >> 2], 1u << ((v & 3u) * 8u));   // ds_add_u32
        }
    }

    // Issuing wave drains TENSORcnt; barrier publishes mask + histogram.
    if (tid < 32) {
#if __has_builtin(__builtin_amdgcn_s_wait_tensorcnt)
        __builtin_amdgcn_s_wait_tensorcnt(0);
#else
        asm volatile("s_wait_tensorcnt 0" ::: "memory");
#endif
    }
    __syncthreads();

    // ---- Phase D: stream logits slice: forbidden mask + rep penalty -----
    const float* lg_row  = logits + (size_t)b * V_ + vLo;
    float*       out_row = out    + (size_t)b * V_ + vLo;
    const float NEG_INF  = -__builtin_inff();

    for (int w = tid; w < VS4; w += NTHREADS) {
        float4   x  = ((const float4*)lg_row)[w];              // global_load_b128
        unsigned fm = ((const unsigned*)fbL)[w];               // 4 mask bytes (from LDS)
        unsigned hw = histW[w];                                // 4 counts     (from LDS)
        if (w + NTHREADS < VS4)
            __builtin_prefetch(lg_row + 4 * (w + NTHREADS), 0, 0);

        float r[4] = {x.x, x.y, x.z, x.w};
        #pragma unroll
        for (int k = 0; k < 4; ++k) {
            float o = ((fm >> (8 * k)) & 0xFFu) ? NEG_INF : r[k];
            unsigned cnt = (hw >> (8 * k)) & 0xFFu;
            if (cnt >= 3u && o != NEG_INF)     // (counts>=MAX_REPS) & (out!=-inf)
                o = apply_penalty(o);
            r[k] = o;
        }
        ((float4*)out_row)[w] = make_float4(r[0], r[1], r[2], r[3]);  // global_store_b128
    }

    // Make Phase-D stores visible to the L2 atomics below.
    __threadfence();
    __syncthreads();

    // ---- Phase E: required-token boost. Boost targets (count==0 & allowed)
    // are disjoint from penalty targets (count>=3) and forbidden (-inf), so
    // applying it last is exact; global f32 atomics handle duplicates like
    // the reference's .at[].add. -----------------------------------------
    if (tid < R_) {
        int tok = required[tid];
        unsigned v = (unsigned)(tok - vLo);
        if (v < (unsigned)VS) {
            unsigned cnt = (histW[v >> 2] >> ((v & 3u) * 8u)) & 0xFFu;
            if (cnt == 0u && fbL[v] == 0)
                atomicAdd(out + (size_t)b * V_ + tok, 5.0f);   // global_atomic_add_f32
        }
    }
}

extern "C" void kernel_launch(void* const* d_in, const int* in_sizes, int n_in,
                              void* d_out, int out_size, void* d_ws, size_t ws_size,
                              hipStream_t stream) {
    (void)in_sizes; (void)n_in; (void)out_size; (void)d_ws; (void)ws_size;
    const float*         logits    = (const float*)d_in[0];
    const int*           generated = (const int*)d_in[1];
    const unsigned char* forbidden = (const unsigned char*)d_in[2];  // jax bool = 1 byte
    const int*           required  = (const int*)d_in[3];
    float*               out       = (float*)d_out;

    dim3 grid(SLICES, B_);
    logit_constraint_kernel<<<grid, NTHREADS, SMEM_BYTES, stream>>>(
        logits, generated, forbidden, required, out);
}